// RPNProposalSSD_8400956031313
// MI455X (gfx1250) — compile-verified
//
#include <hip/hip_runtime.h>
#include <hip/hip_bf16.h>
#include <stdint.h>

// ---------------------------------------------------------------------------
// RPNProposalSSD for MI455X (gfx1250, wave32)
//
// Latency-bound pipeline (~0.1 GFLOP, ~1.5 MB traffic -> far below both the
// 23.3 TB/s HBM roofline and any matrix-op ceiling; WMMA is not applicable).
// MI455X-specific strategy:
//   * 320KB LDS per WGP: entire padded top-k sort (32768 x 8B = 256KB) is
//     resident in ONE workgroup -> zero global round-trips between passes.
//   * Tensor Data Mover: bulk keys[] global->LDS copy via tensor_load_to_lds
//     (TENSORcnt), issued by wave 0 while other waves zero-fill padding.
//   * Async global->LDS gather (ASYNCcnt): the 3000 NMS candidate boxes are
//     gathered straight into LDS with global_load_async_to_lds_b128.
// ---------------------------------------------------------------------------

#define A_ 15
#define H_ 34
#define W_ 34
#define HW_ (H_ * W_)
#define N_ (A_ * HW_)          // 17340 anchors
#define STRIDE_F 8.0f
#define MAX_CAND_ 3000
#define TOP_N_ 300
#define SORT_N_ 32768          // next pow2 >= N_
#define NTHREADS_ 1024         // 32 wave32 waves on one WGP
#define NEG_F (-1e30f)
#define THRESH_ 0.2f
#define MIN_WH_ 6.16056f
#define IOU_TH_ 0.7f

#if defined(__HIP_DEVICE_COMPILE__) && defined(__gfx1250__) && \
    __has_builtin(__builtin_amdgcn_tensor_load_to_lds)
#define HAVE_TDM_ 1
#else
#define HAVE_TDM_ 0
#endif

typedef unsigned int u32x4_ __attribute__((ext_vector_type(4)));
typedef int          i32x4_ __attribute__((ext_vector_type(4)));
typedef int          i32x8_ __attribute__((ext_vector_type(8)));

__device__ __constant__ float c_aw[A_] = {
    9.232984f, 16.0f, 27.712813f, 18.465969f, 32.0f, 55.425626f,
    36.931937f, 64.0f, 110.851252f, 73.863875f, 128.0f, 221.702503f,
    147.72775f, 256.0f, 443.405007f};
__device__ __constant__ float c_ah[A_] = {
    27.72668f, 16.0f, 9.237604f, 55.453359f, 32.0f, 18.475209f,
    110.906719f, 64.0f, 36.950417f, 221.813438f, 128.0f, 73.900834f,
    443.626876f, 256.0f, 147.801669f};
__device__ __constant__ float c_mean[4] = {0.000437f, 0.002586f, -0.123953f, -0.081469f};
__device__ __constant__ float c_std[4]  = {0.12677f, 0.095741f, 0.3173f, 0.281042f};

// ---------------------------------------------------------------------------
// Kernel 1: decode boxes, mask scores, emit monotone 64-bit sort keys.
// Anchor algebra collapses exactly: width=ANCHOR_W[a], height=ANCHOR_H[a],
// ctr_x = (k%W)*8, ctr_y = (k/W)*8 for k = n/A.
// Key = orderable(masked_score)<<32 | (0xFFFFFFFF - n)  => descending u64
// sort reproduces lax.top_k value order + lower-index tie-breaking.
// ---------------------------------------------------------------------------
__global__ void rpn_decode_kernel(const float* __restrict__ cls,
                                  const float* __restrict__ pred,
                                  const int* __restrict__ im_info,
                                  float4* __restrict__ boxes,
                                  float* __restrict__ s0,
                                  float* __restrict__ s1,
                                  unsigned long long* __restrict__ keys) {
    int n = blockIdx.x * blockDim.x + threadIdx.x;
    if (n >= N_) return;

    int a = n % A_;
    int k = n / A_;           // = h*W + w
    int h = k / W_;
    int w = k % W_;

    float ow = (float)im_info[1];
    float oh = (float)im_info[0];

    float ctr_x = (float)w * STRIDE_F;
    float ctr_y = (float)h * STRIDE_F;
    float aw = c_aw[a];
    float ah = c_ah[a];

    int pb = a * 4;
    float d0 = pred[((pb + 0) * H_ + h) * W_ + w] * c_std[0] + c_mean[0];
    float d1 = pred[((pb + 1) * H_ + h) * W_ + w] * c_std[1] + c_mean[1];
    float d2 = pred[((pb + 2) * H_ + h) * W_ + w] * c_std[2] + c_mean[2];
    float d3 = pred[((pb + 3) * H_ + h) * W_ + w] * c_std[3] + c_mean[3];

    float pcx = d0 * aw + ctr_x;
    float pcy = d1 * ah + ctr_y;
    float pw  = expf(d2) * aw;
    float ph  = expf(d3) * ah;

    float x1 = pcx - 0.5f * (pw - 1.0f);
    float y1 = pcy - 0.5f * (ph - 1.0f);
    float x2 = pcx + 0.5f * (pw - 1.0f);
    float y2 = pcy + 0.5f * (ph - 1.0f);

    x1 = fminf(fmaxf(x1, 0.0f), ow - 1.0f);
    x2 = fminf(fmaxf(x2, 0.0f), ow - 1.0f);
    y1 = fminf(fmaxf(y1, 0.0f), oh - 1.0f);
    y2 = fminf(fmaxf(y2, 0.0f), oh - 1.0f);

    float wsx = x2 - x1 + 1.0f;
    float hsx = y2 - y1 + 1.0f;

    float sc0 = cls[(0 * A_ + a) * HW_ + k];
    float sc1 = cls[(1 * A_ + a) * HW_ + k];

    bool keep = (sc1 > THRESH_) && ((wsx >= MIN_WH_) || (hsx >= MIN_WH_));
    float masked = keep ? sc1 : NEG_F;

    unsigned int u = __float_as_uint(masked);
    unsigned int hi = (u & 0x80000000u) ? ~u : (u | 0x80000000u);  // order-preserving
    keys[n] = ((unsigned long long)hi << 32) |
              (unsigned long long)(0xffffffffu - (unsigned int)n);

    boxes[n] = make_float4(x1, y1, x2, y2);
    s0[n] = sc0;
    s1[n] = sc1;
}

// ---------------------------------------------------------------------------
// Kernel 2: single-WGP bitonic top-k sort (256KB LDS) + greedy NMS + emit.
// ---------------------------------------------------------------------------
__global__ void __launch_bounds__(NTHREADS_, 1)
rpn_sort_nms_kernel(const unsigned long long* __restrict__ keys,
                    const float4* __restrict__ boxes,
                    const float* __restrict__ s0,
                    const float* __restrict__ s1,
                    float* __restrict__ out) {
    extern __shared__ unsigned char smem_raw[];
    unsigned long long* skeys = reinterpret_cast<unsigned long long*>(smem_raw);
    const int tid = threadIdx.x;

    // ---- stage keys into LDS ------------------------------------------
    // Padding region [N_, SORT_N_) -> key 0 (sorts to the bottom).
    for (int t = 0; t < (SORT_N_ - N_ + NTHREADS_ - 1) / NTHREADS_; ++t) {
        int e = N_ + tid + t * NTHREADS_;
        if (e < SORT_N_) skeys[e] = 0ull;
    }
#if HAVE_TDM_
    // Bulk contiguous copy keys[0..N) (138,720 B) via the Tensor Data Mover.
    // 1-row tile of 17340 8-byte elements; issued once by wave 0 (TDM is a
    // scalar op, EXEC-independent); other 31 waves zero-fill concurrently.
    if (tid < 32) {
        unsigned long long ga = (unsigned long long)(uintptr_t)keys;
        unsigned int lds_base = (unsigned int)(uintptr_t)skeys;
        u32x4_ g0 = {1u,                                       // count=1 (valid)
                     lds_base,                                 // lds_addr
                     (unsigned int)ga,                         // global_addr lo
                     (unsigned int)(ga >> 32) | (2u << 30)};   // ga hi | type=2
        i32x8_ g1 = {(int)0x00030000,   // data_size=3 (8B); mask/flags=0
                     (int)0x43BC0000,   // tensor_dim0=17340 (bits 79:48, low16)
                     (int)0x00010000,   // tensor_dim0 hi=0 | tensor_dim1=1
                     (int)0x43BC0000,   // tensor_dim1 hi=0 | tile_dim0=17340
                     (int)0x00000001,   // tile_dim1=1, tile_dim2=0
                     (int)17340,        // tensor_dim0_stride lo
                     (int)0x43BC0000,   // dim0_stride hi=0 | dim1_stride lo16
                     0};                // dim1_stride hi
        i32x4_ g2 = {1, 1, 17340, 0};        // tensor_dim2=1, dim3=1, dim2_stride
        i32x4_ g3 = {17340, 1 << 16, 0, 0};  // dim3_stride, tensor_dim4=1
#if __clang_major__ >= 23
        i32x8_ g4 = {0, 0, 0, 0, 0, 0, 0, 0};
        __builtin_amdgcn_tensor_load_to_lds(g0, g1, g2, g3, g4, 0);
#else
        __builtin_amdgcn_tensor_load_to_lds(g0, g1, g2, g3, 0);
#endif
        __builtin_amdgcn_s_wait_tensorcnt(0);
    }
#else
    for (int t = 0; t < SORT_N_ / NTHREADS_; ++t) {
        int e = tid + t * NTHREADS_;
        if (e < N_) skeys[e] = keys[e];
    }
#endif
    __syncthreads();

    // ---- in-LDS bitonic sort, descending ------------------------------
    for (int k2 = 2; k2 <= SORT_N_; k2 <<= 1) {
        for (int j = k2 >> 1; j > 0; j >>= 1) {
            for (int t = 0; t < SORT_N_ / NTHREADS_; ++t) {
                int i = tid + t * NTHREADS_;
                int ixj = i ^ j;
                if (ixj > i) {
                    unsigned long long va = skeys[i];
                    unsigned long long vb = skeys[ixj];
                    bool up = ((i & k2) == 0);
                    if (up ? (va < vb) : (va > vb)) {
                        skeys[i] = vb;
                        skeys[ixj] = va;
                    }
                }
            }
            __syncthreads();
        }
    }

    // ---- extract top-3000 (idx, valid) to registers before LDS reuse ---
    unsigned int ridx[3];
    int rvalid[3];
    for (int t = 0; t < 3; ++t) {
        int c = tid + t * NTHREADS_;
        ridx[t] = 0;
        rvalid[t] = 0;
        if (c < MAX_CAND_) {
            unsigned long long key = skeys[c];
            unsigned int lo = (unsigned int)key;
            unsigned int hi = (unsigned int)(key >> 32);
            unsigned int u = (hi & 0x80000000u) ? (hi & 0x7fffffffu) : ~hi;
            float f = __uint_as_float(u);
            ridx[t] = 0xffffffffu - lo;                 // original anchor index
            rvalid[t] = (f > NEG_F * 0.5f) ? 1 : 0;     // top_vals > NEG*0.5
        }
    }
    __syncthreads();  // done with skeys; repurpose LDS

    // ---- NMS state layout in LDS (all 16B-aligned offsets) -------------
    float4* cbox  = reinterpret_cast<float4*>(smem_raw);            // 3000*16 = 48000
    float*  carea = reinterpret_cast<float*>(smem_raw + 48000);     // 3000*4
    int*    kp    = reinterpret_cast<int*>(smem_raw + 60000);       // 3000*4
    int*    cidx  = reinterpret_cast<int*>(smem_raw + 72000);       // 3000*4
    int*    sel   = reinterpret_cast<int*>(smem_raw + 84000);       // 300*4

    // ---- gather candidate boxes: async global->LDS (gfx1250 ASYNCcnt) --
    for (int t = 0; t < 3; ++t) {
        int c = tid + t * NTHREADS_;
        if (c < MAX_CAND_) {
            kp[c] = rvalid[t];
            cidx[c] = (int)ridx[t];
            unsigned int lds_off = (unsigned int)(uintptr_t)(&cbox[c]);
            const float4* gp = &boxes[ridx[t]];
            asm volatile("global_load_async_to_lds_b128 %0, %1, off"
                         :: "v"(lds_off), "v"(gp)
                         : "memory");
        }
    }
    asm volatile("s_wait_asynccnt 0x0" ::: "memory");
    __syncthreads();

    for (int t = 0; t < 3; ++t) {
        int c = tid + t * NTHREADS_;
        if (c < MAX_CAND_) {
            float4 b = cbox[c];
            carea[c] = (b.z - b.x) * (b.w - b.y);
        }
    }
    __syncthreads();

    // ---- greedy NMS: exact replica of the reference fori_loop ----------
    for (int i = 0; i < MAX_CAND_ - 1; ++i) {
        if (kp[i]) {  // LDS broadcast read, uniform branch
            float4 bi = cbox[i];
            float ai = carea[i];
            for (int j = i + 1 + tid; j < MAX_CAND_; j += NTHREADS_) {
                if (kp[j]) {
                    float4 bj = cbox[j];
                    float iw = fmaxf(fminf(bi.z, bj.z) - fmaxf(bi.x, bj.x), 0.0f);
                    float ih = fmaxf(fminf(bi.w, bj.w) - fmaxf(bi.y, bj.y), 0.0f);
                    float inter = iw * ih;
                    float iou = inter / fmaxf(ai + carea[j] - inter, 1e-12f);
                    if (iou > IOU_TH_) kp[j] = 0;
                }
            }
        }
        __syncthreads();
    }

    // ---- compaction: first 300 surviving indices, fill = MAX_CAND ------
    if (tid == 0) {
        int cnt = 0;
        for (int c = 0; c < MAX_CAND_ && cnt < TOP_N_; ++c)
            if (kp[c]) sel[cnt++] = c;
        for (; cnt < TOP_N_; ++cnt) sel[cnt] = MAX_CAND_;
    }
    __syncthreads();

    // ---- emit: proposals (300x5, col0=0) then out_scores (300x2) -------
    if (tid < TOP_N_) {
        int s = sel[tid];
        float p1 = 0.0f, p2 = 0.0f, p3 = 0.0f, p4 = 0.0f;
        float q0 = 0.0f, q1 = 0.0f;
        if (s < MAX_CAND_) {
            float4 b = cbox[s];
            p1 = b.x; p2 = b.y; p3 = b.z; p4 = b.w;
            int gi = cidx[s];
            q0 = s0[gi];
            q1 = s1[gi];
        }
        out[tid * 5 + 0] = 0.0f;
        out[tid * 5 + 1] = p1;
        out[tid * 5 + 2] = p2;
        out[tid * 5 + 3] = p3;
        out[tid * 5 + 4] = p4;
        out[TOP_N_ * 5 + tid * 2 + 0] = q0;
        out[TOP_N_ * 5 + tid * 2 + 1] = q1;
    }
}

// ---------------------------------------------------------------------------
// Host launch
// ---------------------------------------------------------------------------
extern "C" void kernel_launch(void* const* d_in, const int* in_sizes, int n_in,
                              void* d_out, int out_size, void* d_ws, size_t ws_size,
                              hipStream_t stream) {
    (void)in_sizes; (void)n_in; (void)out_size; (void)ws_size;

    const float* cls     = (const float*)d_in[0];   // (1, 2A, H, W) f32
    const float* pred    = (const float*)d_in[1];   // (1, 4A, H, W) f32
    const int*   im_info = (const int*)d_in[2];     // (2,) i32

    // Workspace layout (~555 KB total):
    //   boxes : float4[N]  @ 0        (277,440 B)
    //   s0    : float [N]  @ 277,440  ( 69,360 B)
    //   s1    : float [N]  @ 346,800  ( 69,360 B)
    //   keys  : u64   [N]  @ 416,160  (138,720 B)  (8B aligned)
    char* ws = (char*)d_ws;
    float4*             boxes = (float4*)(ws);
    float*              s0    = (float*)(ws + (size_t)N_ * 16);
    float*              s1    = (float*)(ws + (size_t)N_ * 16 + (size_t)N_ * 4);
    unsigned long long* keys  = (unsigned long long*)(ws + (size_t)N_ * 16 + (size_t)N_ * 8);

    rpn_decode_kernel<<<(N_ + 255) / 256, 256, 0, stream>>>(
        cls, pred, im_info, boxes, s0, s1, keys);

    // One workgroup, 32 waves, 256 KB dynamic LDS (CDNA5: up to 320 KB/WGP).
    rpn_sort_nms_kernel<<<1, NTHREADS_, SORT_N_ * sizeof(unsigned long long), stream>>>(
        keys, boxes, s0, s1, (float*)d_out);
}